// KmeansAttentionDDP_87608742904390
// MI455X (gfx1250) — compile-verified
//
#include <hip/hip_runtime.h>
#include <hip/hip_bf16.h>

// Problem constants (match reference)
constexpr int B_  = 8;
constexpr int H_  = 8;
constexpr int T_  = 8192;
constexpr int D_  = 64;
constexpr int NC_ = 128;
constexpr int WS_ = 64;           // window size (top-k per cluster)
constexpr int T2_ = 2 * T_;       // concat(q,k) token count per (b,h)
constexpr int BH_ = B_ * H_;
constexpr float EPS_ = 1e-5f;
constexpr float COMMIT_ = 1e-4f;
constexpr float NEGINF_ = -3.402823466e38f;

typedef __attribute__((ext_vector_type(2))) float v2f;
typedef __attribute__((ext_vector_type(8))) float v8f;

// ---------------------------------------------------------------------------
// Kernel 1: per-token inverse L2 norm, centroid argmax, aux-loss accumulation.
// One wave (32 lanes) per token of concat(q,k). 8 waves / block.
// ---------------------------------------------------------------------------
__global__ __launch_bounds__(256)
void norm_argmax_kernel(const float* __restrict__ q, const float* __restrict__ k,
                        const float* __restrict__ means,
                        float* __restrict__ inv_norm, float* __restrict__ aux) {
  const int wave = threadIdx.x >> 5;
  const int lane = threadIdx.x & 31;
  const long long token = (long long)blockIdx.x * 8 + wave;   // over BH*T2
  const int bh = (int)(token / T2_);
  const int t2 = (int)(token % T2_);
  const int h  = bh % H_;

  const float* xrow = (t2 < T_)
      ? (q + ((long long)bh * T_ + t2) * D_)
      : (k + ((long long)bh * T_ + (t2 - T_)) * D_);

  __shared__ float xs[8][64];
  float x0 = xrow[lane];
  float x1 = xrow[lane + 32];
  xs[wave][lane]      = x0;
  xs[wave][lane + 32] = x1;

  float ss = x0 * x0 + x1 * x1;
  #pragma unroll
  for (int m = 16; m > 0; m >>= 1) ss += __shfl_xor(ss, m, 32);
  const float inv = 1.0f / fmaxf(sqrtf(ss), 1e-12f);
  if (lane == 0) inv_norm[token] = inv;
  __syncthreads();

  // Each lane evaluates 4 centroids: c = lane + 32*cc
  const float* mh = means + (long long)h * NC_ * D_;
  float best = NEGINF_;
  int   bidx = 0;
  #pragma unroll
  for (int cc = 0; cc < 4; ++cc) {
    const int c = lane + 32 * cc;
    const float* mr = mh + (long long)c * D_;
    float dot = 0.0f;
    #pragma unroll 8
    for (int d = 0; d < D_; ++d) dot += xs[wave][d] * mr[d];
    const float dist = dot * inv;
    if (dist > best) { best = dist; bidx = c; }     // increasing c => lowest idx on tie
  }
  // cross-lane argmax (lower index wins ties)
  #pragma unroll
  for (int m = 16; m > 0; m >>= 1) {
    const float ov = __shfl_xor(best, m, 32);
    const int   oi = __shfl_xor(bidx, m, 32);
    if (ov > best || (ov == best && oi < bidx)) { best = ov; bidx = oi; }
  }
  // aux contribution: sum_d (xn[d] - means[h,bidx,d])^2
  const float* mb = mh + (long long)bidx * D_;
  const float e0 = xs[wave][lane]      * inv - mb[lane];
  const float e1 = xs[wave][lane + 32] * inv - mb[lane + 32];
  float a = e0 * e0 + e1 * e1;
  #pragma unroll
  for (int m = 16; m > 0; m >>= 1) a += __shfl_xor(a, m, 32);
  if (lane == 0) atomicAdd(aux, a);
}

// ---------------------------------------------------------------------------
// Kernel 2: per-(bh, cluster, q/k-side) top-64 token selection.
// Recomputes the cluster's dist row (8192 dots) into LDS, then 64 iterative
// argmax reductions. Ordering inside the top-64 set does not affect output.
// ---------------------------------------------------------------------------
__global__ __launch_bounds__(256)
void topk_kernel(const float* __restrict__ q, const float* __restrict__ k,
                 const float* __restrict__ means, const float* __restrict__ inv_norm,
                 int* __restrict__ idx_q, int* __restrict__ idx_k) {
  const int gid = blockIdx.x;                 // ((bh*NC + c)*2 + qk)
  const int qk  = gid & 1;
  const int c   = (gid >> 1) % NC_;
  const int bh  = gid >> 1 >> 7;              // (gid/2)/NC
  const int h   = bh % H_;
  const int tid = threadIdx.x;

  __shared__ float msh[64];
  __shared__ float dist[T_];
  __shared__ float rv[256];
  __shared__ int   ri[256];

  if (tid < 64) msh[tid] = means[((long long)h * NC_ + c) * D_ + tid];
  __syncthreads();

  const float* xbase = ((qk == 0) ? q : k) + (long long)bh * T_ * D_;
  const float* invb  = inv_norm + (long long)bh * T2_ + (qk ? T_ : 0);

  for (int t = tid; t < T_; t += 256) {
    const float4* xr = (const float4*)(xbase + (long long)t * D_);
    float dot = 0.0f;
    #pragma unroll
    for (int j = 0; j < 16; ++j) {
      const float4 v4 = xr[j];
      dot += v4.x * msh[4 * j] + v4.y * msh[4 * j + 1] +
             v4.z * msh[4 * j + 2] + v4.w * msh[4 * j + 3];
    }
    dist[t] = dot * invb[t];
  }
  __syncthreads();

  int* outp = ((qk == 0) ? idx_q : idx_k) + ((long long)bh * NC_ + c) * WS_;
  for (int it = 0; it < WS_; ++it) {
    float bv = NEGINF_;
    int   bi = 0;
    for (int t = tid; t < T_; t += 256) {
      const float vv = dist[t];
      if (vv > bv) { bv = vv; bi = t; }        // increasing t => lowest idx on tie
    }
    rv[tid] = bv; ri[tid] = bi;
    __syncthreads();
    #pragma unroll
    for (int s = 128; s > 0; s >>= 1) {
      if (tid < s) {
        const float ov = rv[tid + s];
        const int   oi = ri[tid + s];
        if (ov > rv[tid] || (ov == rv[tid] && oi < ri[tid])) { rv[tid] = ov; ri[tid] = oi; }
      }
      __syncthreads();
    }
    if (tid == 0) { outp[it] = ri[0]; dist[ri[0]] = NEGINF_; }
    __syncthreads();
  }
}

// ---------------------------------------------------------------------------
// Kernel 3: scatter multiplicity counts of q-side indices.
// ---------------------------------------------------------------------------
__global__ __launch_bounds__(256)
void count_kernel(const int* __restrict__ idx_q, float* __restrict__ counts) {
  const long long i = (long long)blockIdx.x * 256 + threadIdx.x;  // BH*NC*WS
  const int bh = (int)(i / (NC_ * WS_));
  const int t  = idx_q[i];
  atomicAdd(&counts[(long long)bh * T_ + t], 1.0f);
}

// ---------------------------------------------------------------------------
// Kernel 4: per-cluster 64x64x64 fp32 attention with V_WMMA_F32_16X16X4_F32.
// Block = 128 threads = 4 waves; wave w owns 16-row M-band w of the 64x64 S/O.
// ---------------------------------------------------------------------------
__global__ __launch_bounds__(128)
void cluster_attn_kernel(const float* __restrict__ q, const float* __restrict__ k,
                         const float* __restrict__ v,
                         const int* __restrict__ idx_q, const int* __restrict__ idx_k,
                         float* __restrict__ out) {
  constexpr int LDW = 68;                       // padded LDS row stride (bank spread)
  const int gid = blockIdx.x;                   // bh*NC + n
  const int n   = gid % NC_;
  const int bh  = gid / NC_;
  const int tid  = threadIdx.x;
  const int wave = tid >> 5;
  const int lane = tid & 31;
  const int hl   = lane & 15;
  const int ksel = (lane >> 4) << 1;            // 0 for lanes 0-15, 2 for 16-31

  __shared__ __align__(16) float qs[64 * LDW];  // q tile, later reused for P
  __shared__ __align__(16) float ks[64 * LDW];
  __shared__ __align__(16) float vs[64 * LDW];

  const int* iq = idx_q + ((long long)bh * NC_ + n) * WS_;
  const int* ik = idx_k + ((long long)bh * NC_ + n) * WS_;
  const float* qb = q + (long long)bh * T_ * D_;
  const float* kb = k + (long long)bh * T_ * D_;
  const float* vb = v + (long long)bh * T_ * D_;

  // Gather 64x64 tiles: each thread loads half a row (8 float4)
  {
    const int row  = tid >> 1;
    const int half = tid & 1;
    const int tq = iq[row];
    const int tk = ik[row];
    const float4* sq = (const float4*)(qb + (long long)tq * D_ + half * 32);
    const float4* sk = (const float4*)(kb + (long long)tk * D_ + half * 32);
    const float4* sv = (const float4*)(vb + (long long)tk * D_ + half * 32);
    float4* dq = (float4*)(qs + row * LDW + half * 32);
    float4* dk = (float4*)(ks + row * LDW + half * 32);
    float4* dv = (float4*)(vs + row * LDW + half * 32);
    #pragma unroll
    for (int j = 0; j < 8; ++j) { dq[j] = sq[j]; dk[j] = sk[j]; dv[j] = sv[j]; }
  }
  __syncthreads();

  const v8f vzero = {0.f, 0.f, 0.f, 0.f, 0.f, 0.f, 0.f, 0.f};
  const int mrow = wave * 16 + hl;              // A-operand row for this lane

  // ---- S = Q * K^T (fp32 WMMA, 16 K-steps of 4) ----
  v8f acc[4];
  #pragma unroll
  for (int ni = 0; ni < 4; ++ni) acc[ni] = vzero;

  #pragma unroll 4
  for (int k0 = 0; k0 < 16; ++k0) {
    const int kc = k0 * 4 + ksel;
    v2f a;
    a.x = qs[mrow * LDW + kc];
    a.y = qs[mrow * LDW + kc + 1];
    #pragma unroll
    for (int ni = 0; ni < 4; ++ni) {
      const int nrow = ni * 16 + hl;
      v2f b;
      b.x = ks[nrow * LDW + kc];
      b.y = ks[nrow * LDW + kc + 1];
      acc[ni] = __builtin_amdgcn_wmma_f32_16x16x4_f32(
          false, a, false, b, (short)0, acc[ni], false, false);
    }
  }

  // ---- row softmax (scale = D^-1/2 = 0.125); rows live in 16-lane halves ----
  const float scale = 0.125f;
  #pragma unroll
  for (int r = 0; r < 8; ++r) {
    float m = NEGINF_;
    #pragma unroll
    for (int ni = 0; ni < 4; ++ni) m = fmaxf(m, acc[ni][r]);
    #pragma unroll
    for (int s = 1; s < 16; s <<= 1) m = fmaxf(m, __shfl_xor(m, s, 32));
    float sum = 0.0f;
    #pragma unroll
    for (int ni = 0; ni < 4; ++ni) {
      const float p = __expf(acc[ni][r] * scale - m * scale);
      acc[ni][r] = p;
      sum += p;
    }
    #pragma unroll
    for (int s = 1; s < 16; s <<= 1) sum += __shfl_xor(sum, s, 32);
    const float rinv = 1.0f / sum;
    #pragma unroll
    for (int ni = 0; ni < 4; ++ni) acc[ni][r] *= rinv;
  }

  // ---- store P into qs (each wave writes only its own 16-row band) ----
  __syncthreads();                                // everyone done reading qs
  #pragma unroll
  for (int r = 0; r < 8; ++r) {
    const int row = wave * 16 + r + ((lane >> 4) << 3);   // +8 for hi half
    #pragma unroll
    for (int ni = 0; ni < 4; ++ni) qs[row * LDW + ni * 16 + hl] = acc[ni][r];
  }
  __syncthreads();

  // ---- O = P * V ----
  v8f oac[4];
  #pragma unroll
  for (int ni = 0; ni < 4; ++ni) oac[ni] = vzero;

  #pragma unroll 4
  for (int k0 = 0; k0 < 16; ++k0) {
    const int kc = k0 * 4 + ksel;
    v2f a;
    a.x = qs[mrow * LDW + kc];
    a.y = qs[mrow * LDW + kc + 1];
    #pragma unroll
    for (int ni = 0; ni < 4; ++ni) {
      const int ncol = ni * 16 + hl;
      v2f b;
      b.x = vs[kc * LDW + ncol];
      b.y = vs[(kc + 1) * LDW + ncol];
      oac[ni] = __builtin_amdgcn_wmma_f32_16x16x4_f32(
          false, a, false, b, (short)0, oac[ni], false, false);
    }
  }

  // ---- scatter O rows to out[bh, iq[row], :] with float atomics ----
  #pragma unroll
  for (int r = 0; r < 8; ++r) {
    const int row = wave * 16 + r + ((lane >> 4) << 3);
    const int tok = iq[row];
    float* orow = out + ((long long)bh * T_ + tok) * D_;
    #pragma unroll
    for (int ni = 0; ni < 4; ++ni) atomicAdd(&orow[ni * 16 + hl], oac[ni][r]);
  }
}

// ---------------------------------------------------------------------------
// Kernel 5: divide by (count + EPS); thread 0 writes aux scalar.
// ---------------------------------------------------------------------------
__global__ __launch_bounds__(256)
void finalize_kernel(float* __restrict__ out, const float* __restrict__ counts,
                     const float* __restrict__ aux) {
  const long long i = (long long)blockIdx.x * 256 + threadIdx.x;  // BH*T*D
  out[i] = out[i] / (counts[i >> 6] + EPS_);
  if (i == 0) {
    const float denom = (float)((long long)B_ * H_ * T2_ * D_);   // 67108864
    out[(long long)BH_ * T_ * D_] = aux[0] * (COMMIT_ / denom);
  }
}

// ---------------------------------------------------------------------------
extern "C" void kernel_launch(void* const* d_in, const int* in_sizes, int n_in,
                              void* d_out, int out_size, void* d_ws, size_t ws_size,
                              hipStream_t stream) {
  const float* q     = (const float*)d_in[0];
  const float* k     = (const float*)d_in[1];
  const float* v     = (const float*)d_in[2];
  const float* means = (const float*)d_in[3];
  float* out = (float*)d_out;
  char*  ws  = (char*)d_ws;

  // Workspace layout (~10.5 MB)
  float* inv_norm = (float*)(ws);                        // BH*T2   = 1,048,576 f
  float* aux      = (float*)(ws + 4194304);              // 16 f (padded)
  int*   idx_q    = (int*)  (ws + 4194368);              // BH*NC*WS = 524,288 i
  int*   idx_k    = (int*)  (ws + 6291520);              // 524,288 i
  float* counts   = (float*)(ws + 8388672);              // BH*T = 524,288 f

  hipMemsetAsync(d_out, 0, (size_t)out_size * sizeof(float), stream);
  hipMemsetAsync(aux, 0, 64, stream);
  hipMemsetAsync(counts, 0, (size_t)BH_ * T_ * sizeof(float), stream);

  // 1) inverse norms + argmax/aux  (BH*T2 tokens, 8 waves/block)
  norm_argmax_kernel<<<dim3((BH_ * T2_) / 8), dim3(256), 0, stream>>>(
      q, k, means, inv_norm, aux);

  // 2) per-cluster top-64 selection for q-side and k-side
  topk_kernel<<<dim3(BH_ * NC_ * 2), dim3(256), 0, stream>>>(
      q, k, means, inv_norm, idx_q, idx_k);

  // 3) multiplicity counts
  count_kernel<<<dim3((BH_ * NC_ * WS_) / 256), dim3(256), 0, stream>>>(idx_q, counts);

  // 4) WMMA cluster attention + scatter
  cluster_attn_kernel<<<dim3(BH_ * NC_), dim3(128), 0, stream>>>(
      q, k, v, idx_q, idx_k, out);

  // 5) normalize + aux scalar
  finalize_kernel<<<dim3((BH_ * T_ * D_) / 256), dim3(256), 0, stream>>>(out, counts, aux);
}